// LastLayer_55362128445545
// MI455X (gfx1250) — compile-verified
//
#include <hip/hip_runtime.h>

// ---------------------------------------------------------------------------
// z = segsum(y[anchor]·w_a) @ W2^T + segsum(sup_x[assign]·w_s) @ W1^T
// N=100000 nodes, E=1.6M edges per set, D=128.
//
// Phase 1: zero two f32 accumulators in d_ws (2 * N * 128 floats = 102.4 MB;
//          fits MI455X's 192MB L2, so the scatter atomics stay L2-resident).
// Phase 2: one wave32 per edge; float4 gather of x[src], scale, 4x
//          global_atomic_add_f32 into agg[dst] (coalesced 512B burst).
// Phase 3: fp32 WMMA GEMM (V_WMMA_F32_16X16X4_F32) computing
//          out = agg1@W1^T + agg2@W2^T, weights staged in padded LDS.
// ---------------------------------------------------------------------------

typedef __attribute__((ext_vector_type(2))) float v2f;
typedef __attribute__((ext_vector_type(8))) float v8f;

#define D 128

__global__ __launch_bounds__(256) void zero_kernel(float4* __restrict__ p, long n4) {
  long i = (long)blockIdx.x * blockDim.x + threadIdx.x;
  if (i < n4) p[i] = make_float4(0.f, 0.f, 0.f, 0.f);
}

// One wave (32 lanes) per edge: lane l handles features [4l, 4l+4).
__global__ __launch_bounds__(256) void scatter_kernel(const float* __restrict__ x,
                                                      const int* __restrict__ eidx,
                                                      const float* __restrict__ ew,
                                                      float* __restrict__ agg,
                                                      int E) {
  int wid  = (int)((blockIdx.x * (unsigned)blockDim.x + threadIdx.x) >> 5);  // wave32
  int lane = threadIdx.x & 31;
  if (wid >= E) return;          // wave-uniform exit
  int   src = eidx[wid];         // row 0 of (2,E) index array
  int   dst = eidx[E + wid];     // row 1
  float w   = ew[wid];
  float4 v  = ((const float4*)(x + (size_t)src * D))[lane];   // 512B/edge gather
  float* ap = agg + (size_t)dst * D + lane * 4;
  atomicAdd(ap + 0, v.x * w);    // global_atomic_add_f32, L2-resident target
  atomicAdd(ap + 1, v.y * w);
  atomicAdd(ap + 2, v.z * w);
  atomicAdd(ap + 3, v.w * w);
}

// out[N,128] = agg1 @ W1^T + agg2 @ W2^T using V_WMMA_F32_16X16X4_F32.
// Block = 256 threads = 8 wave32s; each wave owns one 16-row M-tile and all
// 8 N-tiles (full 128 output features). K loops 0..127 in steps of 4.
__global__ __launch_bounds__(256) void gemm_kernel(const float* __restrict__ agg1,
                                                   const float* __restrict__ W1,
                                                   const float* __restrict__ agg2,
                                                   const float* __restrict__ W2,
                                                   float* __restrict__ out,
                                                   int N) {
  // Padded stride 129: B-frag reads walk rows (stride 129*4B) -> bank step 1,
  // conflict-free across the 16 lanes of each half-wave.
  __shared__ float sW[2][D][D + 1];

  int tid = threadIdx.x;
  for (int i = tid; i < D * D; i += 256) {
    int r = i >> 7, c = i & (D - 1);
    sW[0][r][c] = W1[i];
    sW[1][r][c] = W2[i];
  }
  __syncthreads();

  int wave = tid >> 5;
  int lane = tid & 31;
  int m0   = (blockIdx.x * 8 + wave) * 16;   // N = 100000 is a multiple of 16
  if (m0 >= N) return;                        // wave-uniform (EXEC all-ones for WMMA)

  // A-matrix 16x4 f32 layout (ISA 7.12.2): lanes 0-15 hold K=k0,k0+1 for
  // M=lane; lanes 16-31 hold K=k0+2,k0+3 for M=lane-16.
  int mrow  = lane & 15;
  int khalf = (lane >> 4) * 2;               // 0 or 2
  const float* a1b = agg1 + (size_t)(m0 + mrow) * D + khalf;
  const float* a2b = agg2 + (size_t)(m0 + mrow) * D + khalf;

  v8f acc[8] = {};                           // 8 N-tiles of 16x16 f32 C/D

  for (int k0 = 0; k0 < D; k0 += 4) {
    v2f a1 = *(const v2f*)(a1b + k0);        // float2 per lane
    v2f a2 = *(const v2f*)(a2b + k0);
#pragma unroll
    for (int n = 0; n < 8; ++n) {
      // B 4x16 f32 (K x N), B[k][c] = W[n*16+c][k0+k]; mirrored A layout:
      // lanes 0-15 -> K=k0,k0+1; lanes 16-31 -> K=k0+2,k0+3; column = lane&15.
      int wr = n * 16 + mrow;
      v2f b1, b2;
      b1.x = sW[0][wr][k0 + khalf];
      b1.y = sW[0][wr][k0 + khalf + 1];
      b2.x = sW[1][wr][k0 + khalf];
      b2.y = sW[1][wr][k0 + khalf + 1];
      acc[n] = __builtin_amdgcn_wmma_f32_16x16x4_f32(false, a1, false, b1,
                                                     (short)0, acc[n], false, false);
      acc[n] = __builtin_amdgcn_wmma_f32_16x16x4_f32(false, a2, false, b2,
                                                     (short)0, acc[n], false, false);
    }
  }

  // D layout: VGPR r, lanes 0-15 -> M=r, N=lane; lanes 16-31 -> M=r+8, N=lane-16.
  int rbase = m0 + ((lane >> 4) << 3);
#pragma unroll
  for (int n = 0; n < 8; ++n) {
    int col = n * 16 + mrow;
#pragma unroll
    for (int r = 0; r < 8; ++r) {
      out[(size_t)(rbase + r) * D + col] = acc[n][r];
    }
  }
}

extern "C" void kernel_launch(void* const* d_in, const int* in_sizes, int n_in,
                              void* d_out, int out_size, void* d_ws, size_t ws_size,
                              hipStream_t stream) {
  // setup_inputs order:
  // 0 sup_x [N,128] f32   1 y [N,128] f32
  // 2 assign_index [2,E]  3 assign_weight [E] f32
  // 4 anchor_links [2,E]  5 anchor_weight [E] f32
  // 6 num_nodes (scalar)  7 W1 [128,128] f32   8 W2 [128,128] f32
  const float* sup_x    = (const float*)d_in[0];
  const float* y        = (const float*)d_in[1];
  const int*   assign_i = (const int*)d_in[2];   // harness maps integers -> int32
  const float* assign_w = (const float*)d_in[3];
  const int*   anchor_i = (const int*)d_in[4];
  const float* anchor_w = (const float*)d_in[5];
  const float* W1       = (const float*)d_in[7];
  const float* W2       = (const float*)d_in[8];
  float*       out      = (float*)d_out;

  const int N  = in_sizes[0] / D;   // 100000
  const int E1 = in_sizes[3];       // 1.6M (assign edges)
  const int E2 = in_sizes[5];       // 1.6M (anchor edges)

  float* agg1 = (float*)d_ws;                 // segsum for sup_x/assign
  float* agg2 = agg1 + (size_t)N * D;         // segsum for y/anchor

  // 1) zero both accumulators (float4-vectorized)
  long n4 = (long)N * D * 2 / 4;
  zero_kernel<<<(unsigned)((n4 + 255) / 256), 256, 0, stream>>>((float4*)d_ws, n4);

  // 2) edge scatters (one wave per edge)
  scatter_kernel<<<(unsigned)(((size_t)E1 * 32 + 255) / 256), 256, 0, stream>>>(
      sup_x, assign_i, assign_w, agg1, E1);
  scatter_kernel<<<(unsigned)(((size_t)E2 * 32 + 255) / 256), 256, 0, stream>>>(
      y, anchor_i, anchor_w, agg2, E2);

  // 3) fused dual-GEMM with fp32 WMMA
  int tiles  = (N + 15) / 16;
  int blocks = (tiles + 7) / 8;
  gemm_kernel<<<blocks, 256, 0, stream>>>(agg1, W1, agg2, W2, out, N);
}